// SimulatedQuantumLayer_26216480375363
// MI455X (gfx1250) — compile-verified
//
#include <hip/hip_runtime.h>
#include <math.h>

// CDNA5 (gfx1250) wave32 WMMA vector types
typedef __attribute__((ext_vector_type(2))) float v2f;
typedef __attribute__((ext_vector_type(8))) float v8f;

#define NX        16      // NUM_QUBITS - 1
#define NPH       4096    // 2^NUM_QUBITS_PE
#define NW        240     // (2*ITERATIONS - 1) * (NUM_QUBITS - 1)
#define BATCH_N   16384
#define SCALE_F   4096.0f
#define PI_F      3.14159265358979323846f

// ---------------------------------------------------------------------------
// Kernel 1: angle[b] = frac( sum_j atan(x[b,j]^2) + sum(atan(w^2)) )
// LDS tree reduction for the weight sum (240 elems), float4 row loads.
// ---------------------------------------------------------------------------
__global__ __launch_bounds__(256) void sq_angle_kernel(const float* __restrict__ x,
                                                       const float* __restrict__ w,
                                                       float* __restrict__ angle_out) {
    __shared__ float sred[256];
    const int t = threadIdx.x;

    float wv = 0.0f;
    if (t < NW) {
        float ww = w[t];
        wv = atanf(ww * ww);
    }
    sred[t] = wv;
    __syncthreads();
    #pragma unroll
    for (int s = 128; s > 0; s >>= 1) {
        if (t < s) sred[t] += sred[t + s];
        __syncthreads();
    }
    const float wsum = sred[0];

    const int row = blockIdx.x * 256 + t;                 // 64 blocks * 256 = 16384
    const float4* xr = (const float4*)(x + (size_t)row * NX);
    float acc = wsum;
    #pragma unroll
    for (int i = 0; i < 4; ++i) {
        float4 v = xr[i];                                 // global_load_b128
        acc += atanf(v.x * v.x) + atanf(v.y * v.y)
             + atanf(v.z * v.z) + atanf(v.w * v.w);
    }
    // all terms >= 0, so frac == mod(.,1.0)
    angle_out[row] = acc - floorf(acc);
}

// ---------------------------------------------------------------------------
// Kernel 2: 16x16 diff tiles via V_WMMA_F32_16X16X4_F32, then the Dirichlet
// kernel math per element, streamed out with nontemporal stores.
//
// A (16x4): A[m][0]=angle_m, A[m][1]=1      -> lanes 0-15 {angle,1}, lanes 16-31 {0,0}
// B (4x16): B[0][n]=1,       B[1][n]=-ph_n  -> lanes 0-15 {1,-ph}, lanes 16-31 {0,0}
// D = A*B + 0  ==> D[m][n] = angle_m - phase_n
// C/D layout: VGPR r, lanes 0-15 -> (M=r, N=lane); lanes 16-31 -> (M=8+r, N=lane-16)
//
// Numerator identity: 4096*phases[k] is an exact integer (phases are 12-bit
// binary fractions), so sin(4096*pi*(angle-phase))^2 == sin(4096*pi*angle)^2,
// a per-row constant. Precompute num2[r] = sin^2(4096*pi*angle_row)/4096^2.
// Per element only the denominator sine + rcp remain.
// ---------------------------------------------------------------------------
__global__ __launch_bounds__(256) void sq_probs_kernel(const float* __restrict__ angle,
                                                       const float* __restrict__ phases,
                                                       float* __restrict__ out) {
    const int lane = threadIdx.x & 31;
    const int wave = threadIdx.x >> 5;          // 0..7 (8 waves per block)
    const int row0 = blockIdx.x * 16;           // 1024 blocks * 16 rows = 16384
    const bool lo  = lane < 16;
    const int  n   = lane & 15;

    // A operand: built once per wave (uniform across the whole tile sweep)
    v2f a;
    a.x = lo ? angle[row0 + n] : 0.0f;
    a.y = lo ? 1.0f : 0.0f;

    const int   rowBase    = row0 + ((lane >> 4) << 3);   // lanes>=16 handle M+8
    const float inv_scale2 = 1.0f / (SCALE_F * SCALE_F);

    // Per-row numerator constants: num2[r] = sin^2(4096*pi*angle_{rowBase+r})/4096^2
    float num2[8];
    #pragma unroll
    for (int r = 0; r < 8; ++r) {
        float ang = angle[rowBase + r];         // broadcast load within half-wave
        float u = SCALE_F * ang;                // in [0, 4096)
        u = u - 2.0f * floorf(0.5f * u);        // reduce mod 2 -> [0, 2)
        float s = __sinf(PI_F * u);
        num2[r] = s * s * inv_scale2;
    }

    for (int tile = wave; tile < NPH / 16; tile += 8) {   // 32 tiles per wave
        const int col0 = tile * 16;

        v2f b;
        b.x = lo ? 1.0f : 0.0f;
        b.y = lo ? -phases[col0 + n] : 0.0f;

        v8f c = {};
        // 8 args: (neg_a, A, neg_b, B, c_mod, C, reuse_a, reuse_b)
        v8f d = __builtin_amdgcn_wmma_f32_16x16x4_f32(
            false, a, false, b, (short)0, c, false, false);

        float* op = out + (size_t)rowBase * NPH + (col0 + n);
        #pragma unroll
        for (int r = 0; r < 8; ++r) {
            float diff = d[r];                            // in (-1, 1)
            float den = __sinf(PI_F * diff);
            den = (fabsf(den) < 1e-10f) ? 1e-10f : den;
            float rc = __builtin_amdgcn_rcpf(den);        // v_rcp_f32, no div fixup
            float p = num2[r] * rc * rc;
            __builtin_nontemporal_store(p, op + (size_t)r * NPH);
        }
    }
}

// ---------------------------------------------------------------------------
extern "C" void kernel_launch(void* const* d_in, const int* in_sizes, int n_in,
                              void* d_out, int out_size, void* d_ws, size_t ws_size,
                              hipStream_t stream) {
    (void)in_sizes; (void)n_in; (void)out_size; (void)ws_size;

    const float* x      = (const float*)d_in[0];   // (16384, 16)
    const float* weight = (const float*)d_in[1];   // (240,)
    const float* phases = (const float*)d_in[2];   // (4096,)
    // d_in[3..5] are the compile-time-known scalars; constants baked in.

    float* angle = (float*)d_ws;                   // 16384 floats of scratch
    float* out   = (float*)d_out;                  // (16384, 4096)

    sq_angle_kernel<<<BATCH_N / 256, 256, 0, stream>>>(x, weight, angle);
    sq_probs_kernel<<<BATCH_N / 16, 256, 0, stream>>>(angle, phases, out);
}